// MappingLayer_429496730226
// MI455X (gfx1250) — compile-verified
//
#include <hip/hip_runtime.h>
#include <hip/hip_bf16.h>
#include <math.h>

#define N_SEG 4096
#define EPS 0.1f

typedef __attribute__((ext_vector_type(2))) float v2f;
typedef __attribute__((ext_vector_type(8))) float v8f;

// ---------------------------------------------------------------------------
// Pass 0: init workspace accumulators
// ---------------------------------------------------------------------------
__global__ void init_ws_kernel(unsigned int* __restrict__ segmin,
                               float* __restrict__ denom,
                               float* __restrict__ num) {
    int s = blockIdx.x * blockDim.x + threadIdx.x;
    if (s < N_SEG) {
        segmin[s] = 0x7F800000u;   // +inf bit pattern
        denom[s]  = 0.0f;
        num[s]    = 0.0f;
    }
}

// ---------------------------------------------------------------------------
// Pass 1: per-pair distances via V_WMMA_F32_16X16X4_F32.
// One wave32 handles 16 pairs. diff tile D is 16x64 f32; dist^2[m] = (D D^T)[m][m].
// A-fragment (16x4) and B-fragment (4x16 = D^T chunk) have identical lane
// layouts (lane%16 = M/N, lane/16 = K-half, VGPR = K within half), so the same
// v2f register pair is passed as both A and B.
// ---------------------------------------------------------------------------
__global__ void dist_wmma_kernel(const float* __restrict__ aux,
                                 const float* __restrict__ src,
                                 const int*   __restrict__ aidx,
                                 const int*   __restrict__ sidx,
                                 const int*   __restrict__ seg,
                                 const float* __restrict__ temp_p,
                                 float*        __restrict__ dists,
                                 unsigned int* __restrict__ segmin,
                                 int N) {
    const int lane = threadIdx.x & 31;
    const int wave = (blockIdx.x * (int)blockDim.x + (int)threadIdx.x) >> 5;
    const int base = wave * 16;
    if (base >= N) return;                 // wave-uniform: EXEC stays all-ones below

    const int m = lane & 15;               // matrix row this lane owns (both halves)
    const int h = lane >> 4;               // K-half selector
    int i      = base + m;
    int iclamp = (i < N) ? i : (N - 1);    // branchless clamp keeps EXEC full
    const size_t a_row = (size_t)aidx[iclamp] * 64u;
    const size_t s_row = (size_t)sidx[iclamp] * 64u;

    // Gather 32 floats from each table per lane as 16 float2 chunks:
    // chunk kc covers K = 4*kc + 2*h + {0,1}; lane-halves together cover full row.
    v2f frag[16];
#pragma unroll
    for (int kc = 0; kc < 16; ++kc) {
        int k = kc * 4 + h * 2;
        v2f a = *(const v2f*)(aux + a_row + k);
        v2f s = *(const v2f*)(src + s_row + k);
        frag[kc] = a - s;
    }

    // Accumulate 16x16 Gram matrix of the diff tile: 16 chained f32 WMMAs.
    v8f c = {};
#pragma unroll
    for (int kc = 0; kc < 16; ++kc) {
        c = __builtin_amdgcn_wmma_f32_16x16x4_f32(
                /*neg_a=*/false, frag[kc],
                /*neg_b=*/false, frag[kc],
                /*c_mod=*/(short)0, c,
                /*reuse_a=*/false, /*reuse_b=*/false);
    }

    // Diagonal extraction per the 16x16 f32 C/D layout:
    //   lanes 0..7  hold diag m=lane      in c[lane]
    //   lanes 24..31 hold diag m=lane-16  in c[lane-24]
    bool lowHalf  = (lane < 8);
    bool highHalf = (lane >= 24);
    bool valid    = lowHalf || highHalf;
    int  r        = lowHalf ? lane : (highHalf ? (lane - 24) : 0);
    int  mym      = lowHalf ? lane : (lane - 16);

    float cv = c[0];
    cv = (r == 1) ? c[1] : cv;
    cv = (r == 2) ? c[2] : cv;
    cv = (r == 3) ? c[3] : cv;
    cv = (r == 4) ? c[4] : cv;
    cv = (r == 5) ? c[5] : cv;
    cv = (r == 6) ? c[6] : cv;
    cv = (r == 7) ? c[7] : cv;

    float dist = sqrtf(cv);
    float t2   = dist / (temp_p[0] + EPS);

    if (valid) {
        int gi = base + mym;
        if (gi < N) {
            dists[gi] = dist;
            // t2 >= 0, so uint bit-pattern min == float min
            atomicMin(&segmin[seg[gi]], __float_as_uint(t2));
        }
    }
}

// ---------------------------------------------------------------------------
// Pass 2: exps + segmented sums. seg_ids are sorted, so do a wave-level
// segmented inclusive scan and only run-tail lanes touch global atomics.
// ---------------------------------------------------------------------------
__global__ void softsum_kernel(const float* __restrict__ dists,
                               const int*   __restrict__ seg,
                               const float* __restrict__ temp_p,
                               const unsigned int* __restrict__ segmin,
                               float* __restrict__ denom,
                               float* __restrict__ num,
                               int N) {
    int i    = blockIdx.x * (int)blockDim.x + (int)threadIdx.x;
    int lane = threadIdx.x & 31;
    bool act = (i < N);
    int  iC  = act ? i : (N - 1);

    int   s  = seg[iC];
    float d  = dists[iC];
    float t2 = d / (temp_p[0] + EPS);
    float sm = __uint_as_float(segmin[s]);
    float e  = __expf(t2 - sm);
    float ne = d * e;
    if (!act) { e = 0.0f; ne = 0.0f; }

    // segmented inclusive scan within the wave (keyed by s)
#pragma unroll
    for (int off = 1; off < 32; off <<= 1) {
        float pe = __shfl_up(e, off, 32);
        float pn = __shfl_up(ne, off, 32);
        int   ps = __shfl_up(s, off, 32);
        if (lane >= off && ps == s) { e += pe; ne += pn; }
    }
    int  nextS = __shfl_down(s, 1, 32);
    bool tail  = (lane == 31) || (nextS != s);
    if (tail) {
        atomicAdd(&denom[s], e);
        atomicAdd(&num[s], ne);
    }
}

// ---------------------------------------------------------------------------
// Pass 3: row_losses = num/denom, global min -> scalar
// ---------------------------------------------------------------------------
__global__ void finalize_kernel(const float* __restrict__ denom,
                                const float* __restrict__ num,
                                float* __restrict__ out) {
    __shared__ float red[256];
    float m = INFINITY;
    for (int s = threadIdx.x; s < N_SEG; s += 256) {
        float dn = denom[s];
        if (dn > 0.0f) {
            float l = num[s] / dn;
            m = fminf(m, l);
        }
    }
    red[threadIdx.x] = m;
    __syncthreads();
    for (int w = 128; w > 0; w >>= 1) {
        if ((int)threadIdx.x < w)
            red[threadIdx.x] = fminf(red[threadIdx.x], red[threadIdx.x + w]);
        __syncthreads();
    }
    if (threadIdx.x == 0) out[0] = red[0];
}

// ---------------------------------------------------------------------------
extern "C" void kernel_launch(void* const* d_in, const int* in_sizes, int n_in,
                              void* d_out, int out_size, void* d_ws, size_t ws_size,
                              hipStream_t stream) {
    const float* aux    = (const float*)d_in[0];
    const float* src    = (const float*)d_in[1];
    const float* temp_p = (const float*)d_in[2];
    const int*   aidx   = (const int*)d_in[3];
    const int*   sidx   = (const int*)d_in[4];
    const int*   seg    = (const int*)d_in[5];
    const int    N      = in_sizes[3];

    // workspace layout: dists[N] | segmin[4096] | denom[4096] | num[4096]
    float*        dists  = (float*)d_ws;
    unsigned int* segmin = (unsigned int*)(dists + N);
    float*        denom  = (float*)(segmin + N_SEG);
    float*        num    = denom + N_SEG;
    float*        out    = (float*)d_out;

    init_ws_kernel<<<(N_SEG + 255) / 256, 256, 0, stream>>>(segmin, denom, num);

    int tiles  = (N + 15) / 16;            // 16 pairs per wave32
    int wavesB = 256 / 32;                 // 8 waves per block
    int blocks = (tiles + wavesB - 1) / wavesB;
    dist_wmma_kernel<<<blocks, 256, 0, stream>>>(aux, src, aidx, sidx, seg, temp_p,
                                                 dists, segmin, N);

    softsum_kernel<<<(N + 255) / 256, 256, 0, stream>>>(dists, seg, temp_p, segmin,
                                                        denom, num, N);

    finalize_kernel<<<1, 256, 0, stream>>>(denom, num, out);
}